// DecodingLoss_BCEBased_69561290326260
// MI455X (gfx1250) — compile-verified
//
#include <hip/hip_runtime.h>
#include <math.h>

// Problem constants (match the reference)
constexpr int   B_  = 1024;
constexpr int   N_  = 16384;
constexpr int   M_  = 8192;
constexpr int   K_  = 8;
constexpr int   WC_ = 8;     // check-row width
constexpr int   WO_ = 128;   // observable-row width
constexpr float EPS_  = 1e-6f;
constexpr float BETA_ = 0.5f;
constexpr float LN2_  = 0.69314718055994530942f;

constexpr int THREADS = 256;                 // 8 waves (wave32) per block
constexpr int COPY_ITERS = N_ * 4 / (THREADS * 16); // 64KB row / (256 lanes * 16B) = 16

// One block per batch row b.
// LDS: t_sh holds tanh(llrs[b,:]/2) (64KB), red for block reduction.
__global__ __launch_bounds__(THREADS)
void decode_loss_rows(const float* __restrict__ llrs,
                      const float* __restrict__ syndromes,
                      const float* __restrict__ observables,
                      const int*   __restrict__ chk_idx,
                      const int*   __restrict__ obs_idx,
                      float*       __restrict__ partials)
{
    __shared__ float t_sh[N_];        // 64 KB
    __shared__ float red[THREADS];

    const int b   = blockIdx.x;
    const int tid = threadIdx.x;

    // ---- Stage llrs[b,:] into LDS with CDNA5 async global->LDS copies ----
    // Each async b128 instruction moves 16B per lane straight into LDS
    // (no VGPR round trip); tracked by ASYNCcnt.
    {
        const float* row = llrs + (size_t)b * N_;
        #pragma unroll
        for (int it = 0; it < COPY_ITERS; ++it) {
            const int elem = (it * THREADS + tid) * 4;          // float index, 16B aligned
            unsigned lds_addr   = (unsigned)(size_t)(&t_sh[elem]); // LDS byte address (low 32b)
            const float* gaddr  = row + elem;
            asm volatile("global_load_async_to_lds_b128 %0, %1, off"
                         :: "v"(lds_addr), "v"(gaddr) : "memory");
        }
        asm volatile("s_wait_asynccnt 0x0" ::: "memory");       // wave's copies done
    }
    __syncthreads();                                            // LDS visible to all waves

    // ---- tanh(x/2) in place: tanh(x/2) = 1 - 2/(exp(x)+1) ----
    #pragma unroll
    for (int it = 0; it < COPY_ITERS; ++it) {
        const int i = (it * THREADS + tid) * 4;
        float4 x = *(const float4*)&t_sh[i];
        float4 t;
        t.x = 1.0f - __fdividef(2.0f, __expf(x.x) + 1.0f);
        t.y = 1.0f - __fdividef(2.0f, __expf(x.y) + 1.0f);
        t.z = 1.0f - __fdividef(2.0f, __expf(x.z) + 1.0f);
        t.w = 1.0f - __fdividef(2.0f, __expf(x.w) + 1.0f);
        *(float4*)&t_sh[i] = t;
    }
    __syncthreads();

    const float lo = -1.0f + EPS_, hi = 1.0f - EPS_;
    float acc = 0.0f;

    // ---- Check-row losses: 8-wide gathered products from LDS ----
    // softplus(-pred) + pred*y  ==  ln2 - log(1 + (1-2y)*p)   for y in {0,1}
    {
        const float* srow = syndromes + (size_t)b * M_;
        const int4*  ip   = (const int4*)chk_idx;               // WC_=8 ints = 2x int4 per row
        for (int m = tid; m < M_; m += THREADS) {
            int4 i0 = ip[2 * m + 0];
            int4 i1 = ip[2 * m + 1];
            float p = t_sh[i0.x] * t_sh[i0.y] * t_sh[i0.z] * t_sh[i0.w]
                    * t_sh[i1.x] * t_sh[i1.y] * t_sh[i1.z] * t_sh[i1.w];
            p = fminf(fmaxf(p, lo), hi);
            float s = 1.0f - 2.0f * srow[m];                    // +1 or -1
            acc += LN2_ - __logf(fmaf(s, p, 1.0f));
        }
        acc *= BETA_;
    }

    // ---- Observable-row losses: K_=8 rows of 128-wide products ----
    if (tid < K_) {
        const int*  oi = obs_idx + tid * WO_;
        float p = 1.0f;
        #pragma unroll 4
        for (int j = 0; j < WO_; j += 4) {
            int4 v = *(const int4*)(oi + j);
            p *= t_sh[v.x] * t_sh[v.y] * t_sh[v.z] * t_sh[v.w];
        }
        p = fminf(fmaxf(p, lo), hi);
        float y = observables[(size_t)b * K_ + tid];
        float s = 1.0f - 2.0f * y;
        acc += (1.0f - BETA_) * (LN2_ - __logf(fmaf(s, p, 1.0f)));
    }

    // ---- Deterministic block reduction ----
    red[tid] = acc;
    __syncthreads();
    #pragma unroll
    for (int s = THREADS / 2; s > 0; s >>= 1) {
        if (tid < s) red[tid] += red[tid + s];
        __syncthreads();
    }
    if (tid == 0) partials[b] = red[0] * (1.0f / (float)B_);    // fold the batch mean here
}

// Deterministic final reduction of B_ partials -> scalar.
__global__ __launch_bounds__(THREADS)
void decode_loss_reduce(const float* __restrict__ partials, float* __restrict__ out)
{
    __shared__ float red[THREADS];
    const int tid = threadIdx.x;
    float a = 0.0f;
    #pragma unroll
    for (int i = tid; i < B_; i += THREADS) a += partials[i];
    red[tid] = a;
    __syncthreads();
    #pragma unroll
    for (int s = THREADS / 2; s > 0; s >>= 1) {
        if (tid < s) red[tid] += red[tid + s];
        __syncthreads();
    }
    if (tid == 0) out[0] = red[0];
}

extern "C" void kernel_launch(void* const* d_in, const int* in_sizes, int n_in,
                              void* d_out, int out_size, void* d_ws, size_t ws_size,
                              hipStream_t stream)
{
    (void)in_sizes; (void)n_in; (void)out_size; (void)ws_size;
    const float* llrs        = (const float*)d_in[0];
    const float* syndromes   = (const float*)d_in[1];
    const float* observables = (const float*)d_in[2];
    const int*   chk_idx     = (const int*)d_in[3];
    const int*   obs_idx     = (const int*)d_in[4];
    float* partials = (float*)d_ws;          // B_ floats of scratch
    float* out      = (float*)d_out;         // scalar fp32

    decode_loss_rows<<<B_, THREADS, 0, stream>>>(llrs, syndromes, observables,
                                                 chk_idx, obs_idx, partials);
    decode_loss_reduce<<<1, THREADS, 0, stream>>>(partials, out);
}